// RoiPoolingConv_88012469830149
// MI455X (gfx1250) — compile-verified
//
#include <hip/hip_runtime.h>

#define POOL 7
#define NUM_ROIS 300
#define FH 200
#define FW 200
#define FC 512

// Native 16-byte vector types (clang builtins require native vectors, not
// HIP_vector_type structs).
typedef int   v4i __attribute__((vector_size(16)));
typedef float v4f __attribute__((vector_size(16)));
// Address-space-qualified pointer typedefs (typedefs reliably carry the AS).
typedef __attribute__((address_space(1))) v4i* gbl_v4i_p;  // global (AS1)
typedef __attribute__((address_space(3))) v4i* lds_v4i_p;  // LDS    (AS3)

// CDNA5 async-copy path: global -> LDS DMA tracked by ASYNCcnt.
#if defined(__AMDGCN__) && defined(__gfx1250__) && \
    __has_builtin(__builtin_amdgcn_global_load_async_to_lds_b128)
#define USE_ASYNC 1
#else
#define USE_ASYNC 0
#endif

#if USE_ASYNC
__device__ __forceinline__ void wait_async_le4() {
#if __has_builtin(__builtin_amdgcn_s_wait_asynccnt)
  __builtin_amdgcn_s_wait_asynccnt(4);
#else
  asm volatile("s_wait_asynccnt 4" ::: "memory");
#endif
}
__device__ __forceinline__ void wait_async_le0() {
#if __has_builtin(__builtin_amdgcn_s_wait_asynccnt)
  __builtin_amdgcn_s_wait_asynccnt(0);
#else
  asm volatile("s_wait_asynccnt 0" ::: "memory");
#endif
}
#endif

// One block of 128 threads (4 wave32) per (roi, py). Thread t owns channels
// [4t, 4t+3] -> all global traffic is coalesced b128. Per cell the 4 corner
// channel-vectors (4 x 2KB) are DMA'd into a per-wave LDS double buffer with
// async loads so the gather latency of cell px+1 hides under the blend/store
// of cell px. Each wave consumes only the LDS bytes its own async
// instructions wrote, so no workgroup barriers are required (per-wave
// in-order async completion -> s_wait_asynccnt <=4 covers the current cell
// while the next cell's 4 loads stay in flight).
__global__ __launch_bounds__(128) void roi_pool_bilerp(
    const float* __restrict__ feat,   // (200,200,512) f32
    const int*   __restrict__ rois,   // (300,4) int32: x0,y0,w,h
    float*       __restrict__ out) {  // (300,7,7,512) f32
  const int t   = threadIdx.x;        // 0..127 -> channel group
  const int blk = blockIdx.x;         // 0..2099
  const int r   = blk / POOL;
  const int py  = blk - r * POOL;

  const int x0 = rois[r * 4 + 0];
  const int y0 = rois[r * 4 + 1];
  const int rw = rois[r * 4 + 2];
  const int rh = rois[r * 4 + 3];

  // y interpolation (uniform across block)
  const float stepy = (float)rh / (float)POOL;
  const float sy    = (float)py * stepy;
  const int   y_lo  = (int)sy;                  // sy >= 0
  const float fy    = sy - (float)y_lo;
  const int   y_hi  = min(y_lo + 1, rh - 1);
  const int   rowLo = (y0 + y_lo) * FW;
  const int   rowHi = (y0 + y_hi) * FW;

  // Precompute per-cell corner element offsets (include this thread's channel)
  const float stepx = (float)rw / (float)POOL;
  const int   c     = t * 4;
  int   g00[POOL], g01[POOL], g10[POOL], g11[POOL];
  float fxv[POOL];
#pragma unroll
  for (int px = 0; px < POOL; ++px) {
    const float sx   = (float)px * stepx;
    const int   x_lo = (int)sx;
    fxv[px]          = sx - (float)x_lo;
    const int x_hi   = min(x_lo + 1, rw - 1);
    const int axl    = x0 + x_lo;
    const int axh    = x0 + x_hi;
    g00[px] = (rowLo + axl) * FC + c;
    g01[px] = (rowLo + axh) * FC + c;
    g10[px] = (rowHi + axl) * FC + c;
    g11[px] = (rowHi + axh) * FC + c;
  }

  const int   obase = (r * (POOL * POOL) + py * POOL) * FC + c;
  const float gy    = 1.0f - fy;

#if USE_ASYNC
  // [buf][corner][thread] : 2*4*128*16B = 16 KB per block
  __shared__ v4f smem[2][4][128];
  float* fnc = const_cast<float*>(feat);

  auto issue = [&](int px, int buf) {
    __builtin_amdgcn_global_load_async_to_lds_b128(
        (gbl_v4i_p)(fnc + g00[px]), (lds_v4i_p)&smem[buf][0][t], 0, 0);
    __builtin_amdgcn_global_load_async_to_lds_b128(
        (gbl_v4i_p)(fnc + g01[px]), (lds_v4i_p)&smem[buf][1][t], 0, 0);
    __builtin_amdgcn_global_load_async_to_lds_b128(
        (gbl_v4i_p)(fnc + g10[px]), (lds_v4i_p)&smem[buf][2][t], 0, 0);
    __builtin_amdgcn_global_load_async_to_lds_b128(
        (gbl_v4i_p)(fnc + g11[px]), (lds_v4i_p)&smem[buf][3][t], 0, 0);
  };

  issue(0, 0);  // prime the pipeline (ASYNCcnt = 4)
#pragma unroll
  for (int px = 0; px < POOL; ++px) {
    const int buf = px & 1;
    if (px + 1 < POOL) {
      issue(px + 1, buf ^ 1);  // ASYNCcnt -> 8
      wait_async_le4();        // in-order completion: cell px's 4 are done
    } else {
      wait_async_le0();
    }
    const v4f v00 = smem[buf][0][t];
    const v4f v01 = smem[buf][1][t];
    const v4f v10 = smem[buf][2][t];
    const v4f v11 = smem[buf][3][t];
    const float fx = fxv[px], gx = 1.0f - fx;
    const v4f o = gy * (gx * v00 + fx * v01) + fy * (gx * v10 + fx * v11);
    // Output is write-once/never-read: non-temporal store keeps the 82 MB
    // feature map resident in the 192 MB L2 across all 300 ROIs.
    __builtin_nontemporal_store(o, reinterpret_cast<v4f*>(out + obase + px * FC));
  }
#else
  // Fallback: direct coalesced b128 gathers + L2 prefetch of the next cell.
#pragma unroll
  for (int px = 0; px < POOL; ++px) {
    if (px + 1 < POOL) {
      __builtin_prefetch(feat + g00[px + 1], 0, 0);
      __builtin_prefetch(feat + g01[px + 1], 0, 0);
      __builtin_prefetch(feat + g10[px + 1], 0, 0);
      __builtin_prefetch(feat + g11[px + 1], 0, 0);
    }
    const v4f v00 = *reinterpret_cast<const v4f*>(feat + g00[px]);
    const v4f v01 = *reinterpret_cast<const v4f*>(feat + g01[px]);
    const v4f v10 = *reinterpret_cast<const v4f*>(feat + g10[px]);
    const v4f v11 = *reinterpret_cast<const v4f*>(feat + g11[px]);
    const float fx = fxv[px], gx = 1.0f - fx;
    const v4f o = gy * (gx * v00 + fx * v01) + fy * (gx * v10 + fx * v11);
    __builtin_nontemporal_store(o, reinterpret_cast<v4f*>(out + obase + px * FC));
  }
#endif
}

extern "C" void kernel_launch(void* const* d_in, const int* in_sizes, int n_in,
                              void* d_out, int out_size, void* d_ws, size_t ws_size,
                              hipStream_t stream) {
  (void)in_sizes; (void)n_in; (void)out_size; (void)d_ws; (void)ws_size;
  const float* feat = (const float*)d_in[0];
  const int*   rois = (const int*)d_in[1];
  float*       out  = (float*)d_out;

  dim3 grid(NUM_ROIS * POOL);  // 2100 blocks: one per (roi, pooled-row)
  dim3 block(128);             // 4 wave32; thread t -> channels [4t, 4t+3]
  hipLaunchKernelGGL(roi_pool_bilerp, grid, block, 0, stream, feat, rois, out);
}